// InvertedResidual1_78915729096988
// MI455X (gfx1250) — compile-verified
//
#include <hip/hip_runtime.h>

// ---------------------------------------------------------------------------
// Fused ShuffleNetV2-style block for MI455X (gfx1250, wave32, WMMA).
//   x(64,232,56,56) fp32 -> split -> 1x1conv+BN+PReLU -> dw3x3+BN
//   -> 1x1conv+BN+PReLU -> concat + channel shuffle (g=2)
// One workgroup = one (batch, 4-row stripe). Whole chain stays in LDS.
// GEMMs use V_WMMA_F32_16X16X4_F32 (exact fp32, 2048 FLOP/instr).
// Out-of-range rows/channels are handled by address clamping: the garbage
// they produce only reaches C-fragment elements that are never stored.
// ---------------------------------------------------------------------------

#define BFC     116                 // branch channels
#define CIN     232
#define IMG_W   56
#define IMG_H   56
#define HW      (IMG_W * IMG_H)
#define TH      4                   // interior rows per workgroup
#define HALO_H  (TH + 2)            // 6
#define HALO_W  (IMG_W + 2)         // 58 (cols 0 and 57 are zero padding)
#define NPIX_H  (HALO_H * IMG_W)    // 336 pixels computed by GEMM1
#define NPIX_I  (TH * IMG_W)        // 224 interior pixels
#define NT1     (NPIX_H / 16)       // 21 n-tiles, exact
#define NT2     (NPIX_I / 16)       // 14 n-tiles, exact
#define KSTEPS  (BFC / 4)           // 29, exact (K=4 per WMMA)
#define MT      8                   // 128 padded output channels / 16
#define NTHREADS 256
#define NWAVES  8

typedef float v2f __attribute__((ext_vector_type(2)));
typedef float v4f __attribute__((ext_vector_type(4)));
typedef float v8f __attribute__((ext_vector_type(8)));

// ---- LDS (~326 KB, fits 320 KB WGP budget with 1 WG/WGP) ----
__shared__ __align__(16) float s_w   [BFC * BFC];            // 53.8 KB (w2 then w3)
__shared__ __align__(16) float s_h1  [BFC * HALO_H * HALO_W];// 157.7 KB haloed GEMM1 out
__shared__ __align__(16) float s_h2  [NPIX_I * BFC];         // 101.5 KB dwconv out, [px][ch]
__shared__ __align__(16) float s_wdw [BFC * 9];              // 4.1 KB
__shared__ float s_sc1[BFC], s_bi1[BFC];
__shared__ float s_sc2[BFC], s_bi2[BFC];
__shared__ float s_sc3[BFC], s_bi3[BFC];

__global__ __launch_bounds__(NTHREADS, 1)
void fused_shuffle_block(const float* __restrict__ x,
                         const float* __restrict__ w2,
                         const float* __restrict__ g1, const float* __restrict__ b1,
                         const float* __restrict__ m1, const float* __restrict__ v1,
                         const float* __restrict__ a1p,
                         const float* __restrict__ wdw,
                         const float* __restrict__ g2, const float* __restrict__ b2,
                         const float* __restrict__ m2, const float* __restrict__ v2p,
                         const float* __restrict__ w3,
                         const float* __restrict__ g3, const float* __restrict__ b3,
                         const float* __restrict__ m3, const float* __restrict__ v3p,
                         const float* __restrict__ a2p,
                         float* __restrict__ out)
{
    const int tid  = threadIdx.x;
    const int lane = tid & 31;
    const int wave = tid >> 5;
    const int col  = lane & 15;   // N index inside a 16-wide tile
    const int hi   = lane >> 4;   // lane half: selects K pair / M+8 rows

    const int wg = blockIdx.x;
    const int b  = wg / (IMG_H / TH);
    const int rt = wg % (IMG_H / TH);
    const int y0 = rt * TH;

    // ---------------- phase 0: stage w2, dw weights, BN affine; zero h1 ----
    {
        const v4f* src = (const v4f*)w2;
        v4f* dst = (v4f*)s_w;
        for (int i = tid; i < (BFC * BFC) / 4; i += NTHREADS) dst[i] = src[i];
        const v4f* sdw = (const v4f*)wdw;
        v4f* ddw = (v4f*)s_wdw;
        for (int i = tid; i < (BFC * 9) / 4; i += NTHREADS) ddw[i] = sdw[i];
        if (tid < BFC) {
            float s1 = g1[tid] * rsqrtf(v1[tid] + 1e-5f);
            s_sc1[tid] = s1; s_bi1[tid] = b1[tid] - m1[tid] * s1;
            float s2 = g2[tid] * rsqrtf(v2p[tid] + 1e-5f);
            s_sc2[tid] = s2; s_bi2[tid] = b2[tid] - m2[tid] * s2;
            float s3 = g3[tid] * rsqrtf(v3p[tid] + 1e-5f);
            s_sc3[tid] = s3; s_bi3[tid] = b3[tid] - m3[tid] * s3;
        }
        v4f z = {0.f, 0.f, 0.f, 0.f};
        v4f* h1v = (v4f*)s_h1;
        for (int i = tid; i < (BFC * HALO_H * HALO_W) / 4; i += NTHREADS) h1v[i] = z;
    }
    __syncthreads();

    const float alpha1 = a1p[0];
    const float alpha2 = a2p[0];

    // ---------------- phase 1: GEMM1 (w2 x x2) + BN1 + PReLU -> s_h1 -------
    // A(16x4): lane<16 holds row M=lane {K0,K1}; lane>=16 row M=lane-16 {K2,K3}
    // B(4x16): vgpr0 = rows {K0|K2} striped over lanes, vgpr1 = rows {K1|K3}
    {
        const float* x2base = x + (size_t)(b * CIN + BFC) * HW;
        for (int nt = wave; nt < NT1; nt += NWAVES) {
            const int px = nt * 16 + col;     // 0..335 over the 6x56 halo rows
            const int hy = px / IMG_W;
            const int hx = px % IMG_W;
            const int gy = y0 - 1 + hy;
            const bool vrow = (gy >= 0) && (gy < IMG_H);
            // Clamp the row for address validity; invalid rows produce garbage
            // that only lands in discarded C columns (stores are vrow-guarded).
            const int gyc = gy < 0 ? 0 : (gy >= IMG_H ? IMG_H - 1 : gy);
            const float* bp = x2base + (size_t)gyc * IMG_W + hx + (size_t)(hi * 2) * HW;

            v8f acc[MT];
            #pragma unroll
            for (int mt = 0; mt < MT; ++mt) acc[mt] = (v8f){0,0,0,0,0,0,0,0};

            for (int ks = 0; ks < KSTEPS; ++ks) {
                v2f bf;
                bf.x = bp[(size_t)(ks * 4)     * HW];   // channel c0
                bf.y = bp[(size_t)(ks * 4 + 1) * HW];   // channel c0+1
                const int c0 = ks * 4 + hi * 2;
                #pragma unroll
                for (int mt = 0; mt < MT; ++mt) {
                    // rows >= 116 clamped: their outputs are never stored
                    const int row = (mt * 16 + col < BFC) ? (mt * 16 + col) : (BFC - 1);
                    const float* wp = &s_w[row * BFC + c0];
                    v2f af; af.x = wp[0]; af.y = wp[1];
                    acc[mt] = __builtin_amdgcn_wmma_f32_16x16x4_f32(
                        false, af, false, bf, (short)0, acc[mt], false, false);
                }
            }
            // C/D: vgpr i -> M = i (lanes 0-15) or i+8 (lanes 16-31), N = lane%16
            #pragma unroll
            for (int mt = 0; mt < MT; ++mt) {
                #pragma unroll
                for (int i = 0; i < 8; ++i) {
                    const int ch = mt * 16 + i + hi * 8;
                    if (ch < BFC && vrow) {
                        float v = acc[mt][i] * s_sc1[ch] + s_bi1[ch];
                        v = (v >= 0.f) ? v : alpha1 * v;
                        s_h1[ch * (HALO_H * HALO_W) + hy * HALO_W + (hx + 1)] = v;
                    }
                }
            }
        }
    }
    __syncthreads();

    // ------- phase 2: dw3x3 + BN2 -> s_h2 ; stage w3 ; copy bypass half ----
    {
        const v4f* src = (const v4f*)w3;            // overwrite w2 (GEMM1 done)
        v4f* dst = (v4f*)s_w;
        for (int i = tid; i < (BFC * BFC) / 4; i += NTHREADS) dst[i] = src[i];

        for (int idx = tid; idx < BFC * NPIX_I; idx += NTHREADS) {
            const int ch = idx / NPIX_I;
            const int r  = idx % NPIX_I;        // interior pixel
            const int y  = r / IMG_W;
            const int xx = r % IMG_W;
            const float* hp = &s_h1[ch * (HALO_H * HALO_W) + y * HALO_W + xx];
            const float* wp = &s_wdw[ch * 9];
            float a = 0.f;
            #pragma unroll
            for (int dy = 0; dy < 3; ++dy)
                #pragma unroll
                for (int dx = 0; dx < 3; ++dx)
                    a = fmaf(hp[dy * HALO_W + dx], wp[dy * 3 + dx], a);
            s_h2[r * BFC + ch] = a * s_sc2[ch] + s_bi2[ch];   // pixel-major
        }

        // bypass half: out channel 2*ch = x[:, ch]  (channel shuffle, g=2)
        const float* x1base = x + (size_t)b * CIN * HW;
        for (int i = tid; i < BFC * TH * (IMG_W / 4); i += NTHREADS) {
            const int ch = i / (TH * (IMG_W / 4));
            const int r  = i % (TH * (IMG_W / 4));
            const int y  = r / (IMG_W / 4);
            const int q  = r % (IMG_W / 4);
            const v4f* s = (const v4f*)(x1base + ((size_t)ch * IMG_H + (y0 + y)) * IMG_W) + q;
            v4f* d = (v4f*)(out + ((size_t)(b * CIN + 2 * ch) * IMG_H + (y0 + y)) * IMG_W) + q;
            *d = *s;
        }
    }
    __syncthreads();

    // ---------------- phase 3: GEMM2 (w3 x h2) + BN3 + PReLU -> out --------
    {
        for (int nt = wave; nt < NT2; nt += NWAVES) {
            const int px = nt * 16 + col;     // 0..223 interior
            const float* h2p = &s_h2[px * BFC + hi * 2];   // 8B aligned (even)

            v8f acc[MT];
            #pragma unroll
            for (int mt = 0; mt < MT; ++mt) acc[mt] = (v8f){0,0,0,0,0,0,0,0};

            for (int ks = 0; ks < KSTEPS; ++ks) {
                const v2f bf = *(const v2f*)(h2p + ks * 4);  // ds_load_b64
                const int c0 = ks * 4 + hi * 2;
                #pragma unroll
                for (int mt = 0; mt < MT; ++mt) {
                    const int row = (mt * 16 + col < BFC) ? (mt * 16 + col) : (BFC - 1);
                    const float* wp = &s_w[row * BFC + c0];
                    v2f af; af.x = wp[0]; af.y = wp[1];
                    acc[mt] = __builtin_amdgcn_wmma_f32_16x16x4_f32(
                        false, af, false, bf, (short)0, acc[mt], false, false);
                }
            }
            const int y  = px / IMG_W;
            const int xx = px % IMG_W;
            const int gy = y0 + y;
            #pragma unroll
            for (int mt = 0; mt < MT; ++mt) {
                #pragma unroll
                for (int i = 0; i < 8; ++i) {
                    const int ch = mt * 16 + i + hi * 8;
                    if (ch < BFC) {
                        float v = acc[mt][i] * s_sc3[ch] + s_bi3[ch];
                        v = (v >= 0.f) ? v : alpha2 * v;
                        // channel shuffle: h channel ch -> out channel 2*ch+1
                        out[((size_t)(b * CIN + 2 * ch + 1) * IMG_H + gy) * IMG_W + xx] = v;
                    }
                }
            }
        }
    }
}

extern "C" void kernel_launch(void* const* d_in, const int* in_sizes, int n_in,
                              void* d_out, int out_size, void* d_ws, size_t ws_size,
                              hipStream_t stream) {
    const float* x   = (const float*)d_in[0];
    const float* w2  = (const float*)d_in[1];
    const float* g1  = (const float*)d_in[2];
    const float* b1  = (const float*)d_in[3];
    const float* m1  = (const float*)d_in[4];
    const float* v1  = (const float*)d_in[5];
    const float* a1  = (const float*)d_in[6];
    const float* wdw = (const float*)d_in[7];
    const float* g2  = (const float*)d_in[8];
    const float* b2  = (const float*)d_in[9];
    const float* m2  = (const float*)d_in[10];
    const float* v2p = (const float*)d_in[11];
    const float* w3  = (const float*)d_in[12];
    const float* g3  = (const float*)d_in[13];
    const float* b3  = (const float*)d_in[14];
    const float* m3  = (const float*)d_in[15];
    const float* v3p = (const float*)d_in[16];
    const float* a2  = (const float*)d_in[17];
    float* out = (float*)d_out;

    const int nblocks = 64 * (IMG_H / TH);   // batch x row-stripes = 896
    fused_shuffle_block<<<nblocks, NTHREADS, 0, stream>>>(
        x, w2, g1, b1, m1, v1, a1, wdw, g2, b2, m2, v2p,
        w3, g3, b3, m3, v3p, a2, out);
}